// BysMamba_53223234732595
// MI455X (gfx1250) — compile-verified
//
#include <hip/hip_runtime.h>
#include <hip/hip_bf16.h>
#include <math.h>

// ---- model constants (from reference) ----
#define DIM     256
#define D_STATE 16
#define D_CONV  4
#define D_INNER 512
#define DT_RANK 16
#define BSZ     2
#define SEQ     2048
#define NTOK    (BSZ * SEQ)      // 4096 tokens
#define VOCAB   474
#define N_MAMBA 10
#define DBC_W   (DT_RANK + 2 * D_STATE)   // 48

typedef __attribute__((ext_vector_type(2))) float v2f;
typedef __attribute__((ext_vector_type(8))) float v8f;

enum { EP_STORE = 0, EP_SOFTPLUS_BIAS = 1, EP_ACC_SCALED = 2 };

// ---------------------------------------------------------------------------
// fp32 WMMA GEMM:  C[M,N] (+)= A[M,K] * W[N,K]^T  (weights row-major (out,in)).
// One wave computes a 16 x (NT*16) block; one A fragment feeds NT
// V_WMMA_F32_16X16X4_F32 per K-step. K-loop is software-pipelined one step
// deep: fragments for step k+4 are loaded before the step-k WMMAs issue, so
// the load latency overlaps the matrix pipe instead of stalling on
// s_wait_loadcnt 0.
// RAGGED=1 handles N % 16 != 0 branch-free (clamped row + 0/1 mask multiply)
// so EXEC stays all-ones for every WMMA (ISA requirement).
// Fragment layouts (16x4 f32 A, 4x16 B, 16x16 C):
//   lane m=lane&15, kh=lane>>4 ; a[v]=A[m][2*kh+v] ; b[v]=W[n][2*kh+v]
//   c[r] -> row r+8*kh, col lane&15.
// ---------------------------------------------------------------------------
template <int EPI, int NT, int RAGGED>
__global__ __launch_bounds__(256)
void gemm_wmma_f32(const float* __restrict__ A, int lda,
                   const float* __restrict__ W, int ldw,
                   float* __restrict__ C, int ldc,
                   int M, int N, int K,
                   const float* __restrict__ bias, float scale)
{
    const int mtiles  = (M + 15) >> 4;
    const int ntiles  = (N + 15) >> 4;
    const int ngroups = (ntiles + NT - 1) / NT;

    int wave = (int)((blockIdx.x * blockDim.x + threadIdx.x) >> 5);
    int mt = wave / ngroups;
    int ng = wave - mt * ngroups;
    if (mt >= mtiles) return;                 // wave-uniform exit

    int lane = threadIdx.x & 31;
    int m16  = lane & 15;
    int kh   = lane >> 4;

    int row  = mt * 16 + m16;                 // M is a multiple of 16 everywhere
    int col0 = ng * NT * 16;

    const float* __restrict__ Ap = A + (size_t)row * lda + 2 * kh;
    const float* __restrict__ Wp[NT];
    int   ncol[NT];
    float msk[NT];
#pragma unroll
    for (int j = 0; j < NT; ++j) {
        int n = col0 + j * 16 + m16;
        ncol[j] = n;
        int nc  = RAGGED ? ((n < N) ? n : (N - 1)) : n;   // clamp, in-bounds
        msk[j]  = RAGGED ? ((n < N) ? 1.f : 0.f) : 1.f;
        Wp[j]   = W + (size_t)nc * ldw + 2 * kh;
    }

    v8f acc[NT];
#pragma unroll
    for (int j = 0; j < NT; ++j)
        acc[j] = (v8f){0.f, 0.f, 0.f, 0.f, 0.f, 0.f, 0.f, 0.f};

    // ---- software-pipelined K loop (1 stage) ----
    v2f a_cur, b_cur[NT];
    a_cur[0] = Ap[0];
    a_cur[1] = Ap[1];
#pragma unroll
    for (int j = 0; j < NT; ++j) {
        b_cur[j][0] = Wp[j][0];
        b_cur[j][1] = Wp[j][1];
        if (RAGGED) { b_cur[j][0] *= msk[j]; b_cur[j][1] *= msk[j]; }
    }

    for (int k = 4; k < K; k += 4) {
        v2f a_nxt, b_nxt[NT];
        a_nxt[0] = Ap[k];                      // next-step loads issue first
        a_nxt[1] = Ap[k + 1];
#pragma unroll
        for (int j = 0; j < NT; ++j) {
            b_nxt[j][0] = Wp[j][k];
            b_nxt[j][1] = Wp[j][k + 1];
            if (RAGGED) { b_nxt[j][0] *= msk[j]; b_nxt[j][1] *= msk[j]; }
        }
#pragma unroll
        for (int j = 0; j < NT; ++j)           // overlap with in-flight loads
            acc[j] = __builtin_amdgcn_wmma_f32_16x16x4_f32(
                false, a_cur, false, b_cur[j], (short)0, acc[j], false, false);
        a_cur = a_nxt;
#pragma unroll
        for (int j = 0; j < NT; ++j) b_cur[j] = b_nxt[j];
    }
#pragma unroll
    for (int j = 0; j < NT; ++j)               // epilogue step
        acc[j] = __builtin_amdgcn_wmma_f32_16x16x4_f32(
            false, a_cur, false, b_cur[j], (short)0, acc[j], false, false);

    // ---- epilogue ----
#pragma unroll
    for (int j = 0; j < NT; ++j) {
        int n = ncol[j];
        if (RAGGED && n >= N) continue;
        float bv = (EPI == EP_SOFTPLUS_BIAS) ? bias[n] : 0.f;
#pragma unroll
        for (int r = 0; r < 8; ++r) {
            int rr = mt * 16 + r + 8 * kh;
            size_t off = (size_t)rr * ldc + n;
            float v = acc[j][r];
            if (EPI == EP_STORE) {
                C[off] = v;
            } else if (EPI == EP_SOFTPLUS_BIAS) {
                float t = v + bv;                       // softplus
                C[off] = (t > 20.f) ? t : log1pf(__expf(t));
            } else {                                    // EP_ACC_SCALED
                C[off] += scale * v;
            }
        }
    }
}

// ---------------------------------------------------------------------------
// Front-end: h[bl, o] = 0.5 * (emb[x[bl,1,1]][o] + conv2d_b[o] +
//                              sum_{p,c} emb[x[bl,p]][c] * w2d[o,c,p])
// One block per token; 9x256 embedding tile staged in LDS.
// ---------------------------------------------------------------------------
__global__ __launch_bounds__(256)
void embed_front(const int* __restrict__ x, const float* __restrict__ emb,
                 const float* __restrict__ w2d, const float* __restrict__ b2d,
                 float* __restrict__ h)
{
    __shared__ float tile[9 * DIM];
    int bl = blockIdx.x;
    int c  = threadIdx.x;
#pragma unroll
    for (int p = 0; p < 9; ++p) {
        int tok = x[bl * 9 + p];
        tile[p * DIM + c] = emb[(size_t)tok * DIM + c];
    }
    __syncthreads();
    float center = tile[4 * DIM + c];
    float acc = b2d[c];
    const float* wrow = w2d + (size_t)c * (DIM * 9);
    for (int cc = 0; cc < DIM; ++cc) {
        const float* wp = wrow + cc * 9;
#pragma unroll
        for (int p = 0; p < 9; ++p)
            acc = fmaf(tile[p * DIM + cc], wp[p], acc);
    }
    h[(size_t)bl * DIM + c] = (center + acc) * 0.5f;
}

// ---------------------------------------------------------------------------
// Depthwise causal conv1d (k=4) + bias + SiLU over xz[:, :, 0:512] -> xi.
// rev=1 implements flip(conv(flip(x))) stored at original positions.
// ---------------------------------------------------------------------------
__global__ __launch_bounds__(256)
void conv1d_silu(const float* __restrict__ xz, const float* __restrict__ cw,
                 const float* __restrict__ cb, float* __restrict__ xi, int rev)
{
    int gid = blockIdx.x * blockDim.x + threadIdx.x;   // NTOK * D_INNER
    int d  = gid & (D_INNER - 1);
    int bl = gid >> 9;
    int l  = bl & (SEQ - 1);
    int b  = bl >> 11;
    float acc = cb[d];
#pragma unroll
    for (int t = 0; t < D_CONV; ++t) {
        int ls = rev ? (l + (D_CONV - 1) - t) : (l - (D_CONV - 1) + t);
        if (ls >= 0 && ls < SEQ)
            acc = fmaf(cw[d * D_CONV + t],
                       xz[((size_t)(b * SEQ + ls)) * (2 * D_INNER) + d], acc);
    }
    xi[(size_t)bl * D_INNER + d] = acc / (1.f + __expf(-acc));   // silu
}

// ---------------------------------------------------------------------------
// Selective scan. One lane per (b, d, n): 16-lane groups reduce y = sum_n h*C
// with shfl_xor butterflies. rev=1 scans positions L-1..0. Fuses +D*x and the
// SiLU(z) gate; writes gated y into yg.
// ---------------------------------------------------------------------------
__global__ __launch_bounds__(256)
void scan_kernel(const float* __restrict__ dl, const float* __restrict__ xi,
                 const float* __restrict__ dbc, const float* __restrict__ xz,
                 const float* __restrict__ a_log, const float* __restrict__ dprm,
                 float* __restrict__ yg, int rev)
{
    int gid  = blockIdx.x * blockDim.x + threadIdx.x;  // BSZ*D_INNER*D_STATE
    int n    = gid & (D_STATE - 1);
    int pair = gid >> 4;
    int d    = pair & (D_INNER - 1);
    int b    = pair >> 9;

    float An = -__expf(a_log[d * D_STATE + n]);
    float Dp = dprm[d];
    float h  = 0.f;
    int base = b * SEQ;

    for (int t = 0; t < SEQ; ++t) {
        int l = rev ? (SEQ - 1 - t) : t;
        size_t idx = (size_t)(base + l);
        // prefetch a few steps ahead (predictable stream; hides L2 latency
        // behind the dependent exp/fma chain)
        int tp = t + 8;
        if (tp < SEQ) {
            int lp = rev ? (SEQ - 1 - tp) : tp;
            size_t ip = (size_t)(base + lp);
            __builtin_prefetch(&dbc[ip * DBC_W + DT_RANK], 0, 0);
            __builtin_prefetch(&dl[ip * D_INNER + d], 0, 0);
        }
        float delta = dl[idx * D_INNER + d];
        float xv    = xi[idx * D_INNER + d];
        float Bn    = dbc[idx * DBC_W + DT_RANK + n];
        float Cn    = dbc[idx * DBC_W + DT_RANK + D_STATE + n];
        float dA    = __expf(delta * An);
        h = fmaf(dA, h, delta * Bn * xv);
        float p = h * Cn;
        p += __shfl_xor(p, 1);
        p += __shfl_xor(p, 2);
        p += __shfl_xor(p, 4);
        p += __shfl_xor(p, 8);
        if (n == 0) {
            float z = xz[idx * (2 * D_INNER) + D_INNER + d];
            float y = p + Dp * xv;
            yg[idx * D_INNER + d] = y * (z / (1.f + __expf(-z)));
        }
    }
}

// ---------------------------------------------------------------------------
// Host orchestration
// ---------------------------------------------------------------------------
template <int EPI, int NT, int RAGGED>
static inline void launch_gemm(const float* A, int lda, const float* W, int ldw,
                               float* C, int ldc, int M, int N, int K,
                               const float* bias, float scale, hipStream_t s)
{
    int mtiles  = (M + 15) / 16;
    int ntiles  = (N + 15) / 16;
    int ngroups = (ntiles + NT - 1) / NT;
    int waves   = mtiles * ngroups;
    int blocks  = (waves + 7) / 8;         // 8 waves (256 threads) per block
    gemm_wmma_f32<EPI, NT, RAGGED><<<blocks, 256, 0, s>>>(
        A, lda, W, ldw, C, ldc, M, N, K, bias, scale);
}

extern "C" void kernel_launch(void* const* d_in, const int* in_sizes, int n_in,
                              void* d_out, int out_size, void* d_ws, size_t ws_size,
                              hipStream_t stream)
{
    const int*   x        = (const int*)  d_in[0];
    const float* emb      = (const float*)d_in[1];
    const float* conv2d_w = (const float*)d_in[2];
    const float* conv2d_b = (const float*)d_in[3];
    const float* w_in     = (const float*)d_in[4];
    const float* conv_w   = (const float*)d_in[5];
    const float* conv_b   = (const float*)d_in[6];
    const float* w_x      = (const float*)d_in[7];
    const float* w_dt     = (const float*)d_in[8];
    const float* b_dt     = (const float*)d_in[9];
    const float* a_log    = (const float*)d_in[10];
    const float* d_param  = (const float*)d_in[11];
    const float* w_out    = (const float*)d_in[12];
    const float* lm_w     = (const float*)d_in[13];
    float* out = (float*)d_out;

    // workspace carve-up (floats)
    float* ws  = (float*)d_ws;
    float* h   = ws;  ws += (size_t)NTOK * DIM;              // residual stream
    float* xz  = ws;  ws += (size_t)NTOK * 2 * D_INNER;      // in-proj output
    float* xi  = ws;  ws += (size_t)NTOK * D_INNER;          // conv+silu output
    float* dbc = ws;  ws += (size_t)NTOK * DBC_W;            // dt|B|C
    float* dl  = ws;  ws += (size_t)NTOK * D_INNER;          // delta (softplus)
    float* yg  = ws;  ws += (size_t)NTOK * D_INNER;          // gated scan output

    // front-end: embedding + 3x3 patch conv
    embed_front<<<NTOK, 256, 0, stream>>>(x, emb, conv2d_w, conv2d_b, h);

    const int conv_blocks = (NTOK * D_INNER) / 256;
    const int scan_blocks = (BSZ * D_INNER * D_STATE) / 256;

    for (int layer = 0; layer < N_MAMBA; ++layer) {
        bool bidir  = (layer == 0) || (layer == N_MAMBA - 1);
        float scale = bidir ? 0.5f : 1.0f;
        int   ndir  = bidir ? 2 : 1;

        // xz = h @ w_in[layer]^T   (direction independent; computed once)
        launch_gemm<EP_STORE, 4, 0>(h, DIM,
                    w_in + (size_t)layer * 2 * D_INNER * DIM, DIM,
                    xz, 2 * D_INNER, NTOK, 2 * D_INNER, DIM, nullptr, 0.f, stream);

        for (int rev = 0; rev < ndir; ++rev) {
            conv1d_silu<<<conv_blocks, 256, 0, stream>>>(
                xz, conv_w + (size_t)layer * D_INNER * D_CONV,
                conv_b + (size_t)layer * D_INNER, xi, rev);

            // dbc = xi @ w_x[layer]^T      (4096 x 48, 3 N-tiles per wave)
            launch_gemm<EP_STORE, 3, 0>(xi, D_INNER,
                        w_x + (size_t)layer * DBC_W * D_INNER, D_INNER,
                        dbc, DBC_W, NTOK, DBC_W, D_INNER, nullptr, 0.f, stream);

            // delta = softplus(dt @ w_dt^T + b_dt)   (A = dbc[:, :16], lda=48)
            launch_gemm<EP_SOFTPLUS_BIAS, 4, 0>(dbc, DBC_W,
                        w_dt + (size_t)layer * D_INNER * DT_RANK, DT_RANK,
                        dl, D_INNER, NTOK, D_INNER, DT_RANK,
                        b_dt + (size_t)layer * D_INNER, 0.f, stream);

            scan_kernel<<<scan_blocks, 256, 0, stream>>>(
                dl, xi, dbc, xz,
                a_log + (size_t)layer * D_INNER * D_STATE,
                d_param + (size_t)layer * D_INNER, yg, rev);

            // h += scale * (yg @ w_out[layer]^T)   (fused residual)
            launch_gemm<EP_ACC_SCALED, 4, 0>(yg, D_INNER,
                        w_out + (size_t)layer * DIM * D_INNER, D_INNER,
                        h, DIM, NTOK, DIM, D_INNER, nullptr, scale, stream);
        }
    }

    // logits = h @ lm_head^T   (N = 474 -> 30 tiles, pairs, ragged tail masked)
    launch_gemm<EP_STORE, 2, 1>(h, DIM, lm_w, DIM, out, VOCAB,
                NTOK, VOCAB, DIM, nullptr, 0.f, stream);
}